// NGramRepeatBlock_17076789969164
// MI455X (gfx1250) — compile-verified
//
#include <hip/hip_runtime.h>
#include <hip/hip_bf16.h>

#define AS1 __attribute__((address_space(1)))
#define AS3 __attribute__((address_space(3)))

typedef float v4f __attribute__((vector_size(16)));
typedef int   v4i __attribute__((vector_size(16)));

#if __has_builtin(__builtin_amdgcn_global_load_async_to_lds_b128) && \
    __has_builtin(__builtin_amdgcn_global_load_async_to_lds_b32) &&  \
    __has_builtin(__builtin_amdgcn_s_wait_asynccnt)
#define USE_ASYNC_LDS 1
#else
#define USE_ASYNC_LDS 0
#endif

#define MAX_LDS_TOKENS 8192  // 32 KB of the 320 KB/WGP LDS

// ---------------------------------------------------------------------------
// Kernel 1: streaming copy lprobs -> out. 206 MB total traffic; b128 NT
// loads/stores, grid-stride so HBM stays saturated. ~8.9 us floor @ 23.3 TB/s.
// ---------------------------------------------------------------------------
__global__ void ngram_copy_kernel(const float* __restrict__ src,
                                  float* __restrict__ dst,
                                  long long n) {
    const long long tid    = (long long)blockIdx.x * blockDim.x + threadIdx.x;
    const long long stride = (long long)gridDim.x * blockDim.x;
    const long long n4     = n >> 2;

    const v4f* __restrict__ s4 = (const v4f*)src;
    v4f* __restrict__ d4       = (v4f*)dst;

    for (long long i = tid; i < n4; i += stride) {
        v4f v = __builtin_nontemporal_load(s4 + i);
        __builtin_nontemporal_store(v, d4 + i);
    }
    // scalar tail (n may not be a multiple of 4 in general)
    for (long long i = (n4 << 2) + tid; i < n; i += stride) {
        dst[i] = src[i];
    }
}

// ---------------------------------------------------------------------------
// Kernel 2: per-row n-gram scan + sparse -inf scatter into out.
// One block per row; tokens[row, 0..step] staged into LDS via the gfx1250
// async global->LDS path (ASYNCcnt), then threads scan windows from LDS.
// ---------------------------------------------------------------------------
__global__ void ngram_ban_kernel(const int* __restrict__ tokens,
                                 float* __restrict__ out,
                                 const int* __restrict__ bsz_p,
                                 const int* __restrict__ step_p,
                                 const int* __restrict__ beam_p,
                                 const int* __restrict__ n_p,
                                 long long tokens_elems,
                                 long long lprobs_elems) {
    const int bsz  = *bsz_p;
    const int step = *step_p;
    const int beam = *beam_p;
    const int n    = *n_p;

    const int R = bsz * beam;
    if (R <= 0) return;
    const long long S = tokens_elems / R;
    const long long V = lprobs_elems / R;

    const int num_windows = step - n + 2;   // starts: 0 .. num_windows-1
    if (num_windows <= 0 || n < 1) return;
    const int count     = step + 1;         // tokens used: indices 0..step
    const int last_base = step - n + 2;     // first index of the trailing (n-1)-gram
    const float NEG_INF = -__builtin_inff();

    __shared__ int s_tok[MAX_LDS_TOKENS];

    for (int row = blockIdx.x; row < R; row += gridDim.x) {
        const int* __restrict__ trow = tokens + (long long)row * S;
        float* __restrict__ orow     = out + (long long)row * V;

        if (count <= MAX_LDS_TOKENS) {
#if USE_ASYNC_LDS
            {
                AS1 v4i* g128 = (AS1 v4i*)trow;   // global src, 16B chunks
                AS3 v4i* l128 = (AS3 v4i*)s_tok;  // LDS dst
                const int nvec = count >> 2;
                for (int i = threadIdx.x; i < nvec; i += (int)blockDim.x) {
                    __builtin_amdgcn_global_load_async_to_lds_b128(
                        g128 + i, l128 + i, 0, 0);
                }
                AS1 int* g32 = (AS1 int*)trow;
                AS3 int* l32 = (AS3 int*)s_tok;
                for (int i = (nvec << 2) + threadIdx.x; i < count; i += (int)blockDim.x) {
                    __builtin_amdgcn_global_load_async_to_lds_b32(
                        g32 + i, l32 + i, 0, 0);
                }
                __builtin_amdgcn_s_wait_asynccnt(0);  // my async writes landed in LDS
                __syncthreads();                      // everyone's writes visible
            }
#else
            for (int i = threadIdx.x; i < count; i += (int)blockDim.x) {
                s_tok[i] = trow[i];
            }
            __syncthreads();
#endif
            for (int w = threadIdx.x; w < num_windows; w += (int)blockDim.x) {
                bool m = true;
                for (int j = 0; j < n - 1; ++j) {
                    m &= (s_tok[w + j] == s_tok[last_base + j]);
                }
                if (m) {
                    const int banned = s_tok[w + n - 1];
                    if (banned >= 0 && banned < (int)V) {
                        orow[banned] = NEG_INF;  // idempotent; races benign
                    }
                }
            }
            __syncthreads();  // protect LDS before next row's staging
        } else {
            // Fallback for very long sequences: scan straight from global (L2-resident).
            for (int w = threadIdx.x; w < num_windows; w += (int)blockDim.x) {
                bool m = true;
                for (int j = 0; j < n - 1; ++j) {
                    m &= (trow[w + j] == trow[last_base + j]);
                }
                if (m) {
                    const int banned = trow[w + n - 1];
                    if (banned >= 0 && banned < (int)V) {
                        orow[banned] = NEG_INF;
                    }
                }
            }
        }
    }
}

extern "C" void kernel_launch(void* const* d_in, const int* in_sizes, int n_in,
                              void* d_out, int out_size, void* d_ws, size_t ws_size,
                              hipStream_t stream) {
    const int*   tokens = (const int*)d_in[0];
    const float* lprobs = (const float*)d_in[1];
    const int*   bsz_p  = (const int*)d_in[2];
    const int*   step_p = (const int*)d_in[3];
    const int*   beam_p = (const int*)d_in[4];
    const int*   n_p    = (const int*)d_in[5];
    float*       out    = (float*)d_out;

    const long long n_elems = (long long)out_size;

    // Copy: enough blocks to keep HBM saturated with b128 NT traffic.
    const int threads = 256;
    long long need = ((n_elems >> 2) + threads - 1) / threads;
    int blocks = (int)(need < 1 ? 1 : (need > 8192 ? 8192 : need));
    ngram_copy_kernel<<<blocks, threads, 0, stream>>>(lprobs, out, n_elems);

    // Ban scatter: grid-stride over rows (R is only known on-device; 1024
    // blocks covers any realistic R, extras exit immediately).
    ngram_ban_kernel<<<1024, 256, 0, stream>>>(
        tokens, out, bsz_p, step_p, beam_p, n_p,
        (long long)in_sizes[0], (long long)in_sizes[1]);
}